// RnnModel_12137577579212
// MI455X (gfx1250) — compile-verified
//
#include <hip/hip_runtime.h>
#include <math.h>

// RNN forward on MI455X (gfx1250), fp32 end-to-end via V_WMMA_F32_16X16X4_F32.
//
// Phase 1: x_proj = input @ W_ih^T + b_ih + b_hh   (32768 x 1024 x 1024 GEMM)
//          -> written straight into d_out (scan overwrites it in place).
//          128x64 tile / 256-thread block; K staging is double-buffered with
//          GLOBAL_LOAD_ASYNC_TO_LDS_B128 (ASYNCcnt) when the toolchain
//          exposes it, else a synchronous VGPR-staged fallback.
// Phase 2: persistent-kernel scan, h_s = tanh(xp_s + h_{s-1} @ W_hh^T),
//          W_hh slice resident in LDS (64 KB), steps separated by a global
//          arrive/spin barrier in d_ws.

#define SEQ    512
#define BATCH  64
#define INDIM  1024
#define HIDDIM 1024

typedef __attribute__((ext_vector_type(2))) float v2f;
typedef __attribute__((ext_vector_type(8))) float v8f;

#if __has_builtin(__builtin_amdgcn_global_load_async_to_lds_b128) && \
    __has_builtin(__builtin_amdgcn_s_wait_asynccnt)
#define HAVE_ASYNC_LDS 1
#else
#define HAVE_ASYNC_LDS 0
#endif

#if HAVE_ASYNC_LDS
// The builtin's exact parameter types (from the clang diagnostic):
//   (v4i addrspace(1)* src, v4i addrspace(3)* dst, imm int offset, imm int cpol)
typedef int gcc_v4i __attribute__((vector_size(16)));
typedef __attribute__((address_space(1))) gcc_v4i* gbl_v4i_ptr;
typedef __attribute__((address_space(3))) gcc_v4i* lds_v4i_ptr;
// generic->LDS: low 32 bits of a generic LDS address are the LDS offset.
static __device__ __forceinline__ lds_v4i_ptr lds_cast(void* p) {
    return (lds_v4i_ptr)(unsigned int)(unsigned long long)p;
}
static __device__ __forceinline__ gbl_v4i_ptr gbl_cast(const void* p) {
    return (gbl_v4i_ptr)(unsigned long long)p;
}
#endif

// D = A(16x4) * B(4x16) + C, fp32 exact.
// A frag: lane = kh*16+m, a[r] = A[m][2*kh+r]
// B frag: lane = kh*16+n, b[r] = B[2*kh+r][n]
// C/D   : VGPR j holds row j (lanes 0-15) / row j+8 (lanes 16-31), col = lane&15
static __device__ __forceinline__ v8f wmma_f32_16x16x4(v2f a, v2f b, v8f c) {
    return __builtin_amdgcn_wmma_f32_16x16x4_f32(
        /*neg_a=*/false, a, /*neg_b=*/false, b,
        /*c_mod=*/(short)0, c, /*reuse_a=*/false, /*reuse_b=*/false);
}

// ---------------------------------------------------------------------------
// Phase 1: x_proj GEMM.  Block = 256 threads (8 waves), tile 128(M) x 64(N).
// Wave w computes rows [w*16, w*16+16) x all 64 N (4 accumulators).
// LDS layouts are byte-exact copies of global rows (async-copyable):
//   ldsA[r*36 + k]  (A rows, pad 4 -> float4-aligned stores, even frag offsets)
//   ldsB[n*34 + k]  (W rows, pad 2 -> even frag offsets, conflict-free b64)
// ---------------------------------------------------------------------------
#define K_CHUNK 32
#define LDA_STRIDE 36
#define LDB_STRIDE 34

__global__ __launch_bounds__(256) void xproj_kernel(
    const float* __restrict__ input,   // [SEQ*BATCH][INDIM]
    const float* __restrict__ W_ih,    // [HIDDIM][INDIM]
    const float* __restrict__ b_ih,    // [HIDDIM]
    const float* __restrict__ b_hh,    // [HIDDIM]
    float* __restrict__ out,           // [SEQ*BATCH][HIDDIM]
    unsigned int* __restrict__ barrier)
{
    __shared__ float ldsA[2][128 * LDA_STRIDE];   // 2 x 18432 B
    __shared__ float ldsB[2][64 * LDB_STRIDE];    // 2 x  8704 B

    const int tid    = threadIdx.x;
    const int wave   = tid >> 5;      // 0..7
    const int lane   = tid & 31;
    const int laneLo = lane & 15;
    const int kh     = lane >> 4;     // K-pair select within a fragment
    const int m0     = blockIdx.y * 128;  // row block in [0, 32768)
    const int n0     = blockIdx.x * 64;   // col block in [0, 1024)

    // Reset the scan barrier for this call (stream-ordered before the scan).
    if (blockIdx.x == 0 && blockIdx.y == 0 && tid == 0) *barrier = 0u;

    // thread -> (row, col4) mapping for staging (16B per thread-slot)
    const int ar[4] = { (tid + 0 * 256) >> 3, (tid + 1 * 256) >> 3,
                        (tid + 2 * 256) >> 3, (tid + 3 * 256) >> 3 };
    const int ac4   = (tid & 7) << 2;
    const int br[2] = { (tid + 0 * 256) >> 3, (tid + 1 * 256) >> 3 };

    // Stage one K-chunk into LDS buffer `buf`.
    auto stage = [&](int k0, int buf) {
#if HAVE_ASYNC_LDS
#pragma unroll
        for (int j = 0; j < 4; ++j)
            __builtin_amdgcn_global_load_async_to_lds_b128(
                gbl_cast(&input[(size_t)(m0 + ar[j]) * INDIM + k0 + ac4]),
                lds_cast(&ldsA[buf][ar[j] * LDA_STRIDE + ac4]), 0, 0);
#pragma unroll
        for (int j = 0; j < 2; ++j)
            __builtin_amdgcn_global_load_async_to_lds_b128(
                gbl_cast(&W_ih[(size_t)(n0 + br[j]) * INDIM + k0 + ac4]),
                lds_cast(&ldsB[buf][br[j] * LDB_STRIDE + ac4]), 0, 0);
#else
#pragma unroll
        for (int j = 0; j < 4; ++j) {
            const float4 v = *reinterpret_cast<const float4*>(
                &input[(size_t)(m0 + ar[j]) * INDIM + k0 + ac4]);
            *reinterpret_cast<float4*>(&ldsA[buf][ar[j] * LDA_STRIDE + ac4]) = v;
        }
#pragma unroll
        for (int j = 0; j < 2; ++j) {
            const float4 v = *reinterpret_cast<const float4*>(
                &W_ih[(size_t)(n0 + br[j]) * INDIM + k0 + ac4]);
            // stride 34 is 2 mod 4 -> float4 store may be 8B-misaligned; use 2x b64
            v2f lo; lo.x = v.x; lo.y = v.y;
            v2f hi; hi.x = v.z; hi.y = v.w;
            *reinterpret_cast<v2f*>(&ldsB[buf][br[j] * LDB_STRIDE + ac4 + 0]) = lo;
            *reinterpret_cast<v2f*>(&ldsB[buf][br[j] * LDB_STRIDE + ac4 + 2]) = hi;
        }
#endif
    };
    auto wait_stage = [&]() {
#if HAVE_ASYNC_LDS
        __builtin_amdgcn_s_wait_asynccnt(0);
#endif
    };

    // Init accumulators with the fused bias (same col for all 8 rows/lane).
    v8f acc[4];
#pragma unroll
    for (int nt = 0; nt < 4; ++nt) {
        const float bv = b_ih[n0 + nt * 16 + laneLo] + b_hh[n0 + nt * 16 + laneLo];
#pragma unroll
        for (int j = 0; j < 8; ++j) acc[nt][j] = bv;
    }

    const int mw = wave * 16;

    // Prologue: stage chunk 0, wait, barrier.
    stage(0, 0);
    wait_stage();
    __syncthreads();

    int cur = 0;
    for (int k0 = 0; k0 < INDIM; k0 += K_CHUNK) {
        // Kick off the next chunk into the other buffer (overlaps compute).
        if (k0 + K_CHUNK < INDIM) stage(k0 + K_CHUNK, cur ^ 1);

        // ---- compute 8 k-quads from LDS buffer `cur` ----
#pragma unroll
        for (int kc = 0; kc < K_CHUNK; kc += 4) {
            const v2f a = *reinterpret_cast<const v2f*>(
                &ldsA[cur][(mw + laneLo) * LDA_STRIDE + kc + 2 * kh]);
#pragma unroll
            for (int nt = 0; nt < 4; ++nt) {
                const v2f b = *reinterpret_cast<const v2f*>(
                    &ldsB[cur][(nt * 16 + laneLo) * LDB_STRIDE + kc + 2 * kh]);
                acc[nt] = wmma_f32_16x16x4(a, b, acc[nt]);
            }
        }

        // All async writes (ours) done + everyone finished reading `cur`.
        wait_stage();
        __syncthreads();
        cur ^= 1;
    }

    // Store: row = m0 + mw + kh*8 + j, col = n0 + nt*16 + laneLo.
    const int rowBase = m0 + mw + kh * 8;
#pragma unroll
    for (int nt = 0; nt < 4; ++nt)
#pragma unroll
        for (int j = 0; j < 8; ++j)
            out[(size_t)(rowBase + j) * HIDDIM + n0 + nt * 16 + laneLo] = acc[nt][j];
}

// ---------------------------------------------------------------------------
// Phase 2: persistent recurrent scan. 64 blocks, each owns 16 hidden columns.
// ldsW pair-interleaved: element [p*16+n] = {W_hh[n0+n][2p], W_hh[n0+n][2p+1]}
// (exactly 64 KB; pair-row stride 16*8B ≡ 32 float-banks so the two lane
// halves hit disjoint bank sets -> conflict-free b64 fragment loads).
// ---------------------------------------------------------------------------
__global__ __launch_bounds__(128) void rnn_scan_kernel(
    const float* __restrict__ W_hh,    // [HIDDIM][HIDDIM]
    float* __restrict__ out,           // [SEQ][BATCH][HIDDIM], holds x_proj on entry
    unsigned int* __restrict__ barrier)
{
    __shared__ v2f ldsW[(HIDDIM / 2) * 16];   // 65536 bytes

    const int tid    = threadIdx.x;
    const int wave   = tid >> 5;
    const int lane   = tid & 31;
    const int laneLo = lane & 15;
    const int kh     = lane >> 4;
    const int n0     = blockIdx.x * 16;
    const unsigned nblocks = gridDim.x;

    // One-time weight preload: W_hh rows n0..n0+15, pair-interleaved k-major.
    for (int i = tid; i < 16 * (HIDDIM / 4); i += 128) {
        const int n  = i >> 8;          // 0..15
        const int c4 = (i & 255) << 2;  // 0..1020
        const float4 v = *reinterpret_cast<const float4*>(
            &W_hh[(size_t)(n0 + n) * HIDDIM + c4]);
        v2f lo; lo.x = v.x; lo.y = v.y;
        v2f hi; hi.x = v.z; hi.y = v.w;
        ldsW[((c4 >> 1) + 0) * 16 + n] = lo;
        ldsW[((c4 >> 1) + 1) * 16 + n] = hi;
    }
    __syncthreads();

    const int mrow    = wave * 16 + laneLo;   // h row this lane streams for A
    const int rowBase = wave * 16 + kh * 8;   // C/D row base for this lane

    for (int s = 0; s < SEQ; ++s) {
        float* xp = out + (size_t)s * BATCH * HIDDIM;

        if (s == 0) {
            // h0 = 0  ->  h1 = tanh(xp) elementwise on this block's 64x16 slice.
            for (int i = tid; i < 64 * 16; i += 128) {
                const int r = i >> 4, c = i & 15;
                const size_t idx = (size_t)r * HIDDIM + n0 + c;
                xp[idx] = tanhf(xp[idx]);
            }
        } else {
            const float* hprev = out + (size_t)(s - 1) * BATCH * HIDDIM;
            // 4 independent accumulator chains to hide the WMMA RAW latency.
            v8f acc0 = {}, acc1 = {}, acc2 = {}, acc3 = {};
#pragma unroll 2
            for (int k = 0; k < HIDDIM; k += 16) {
                const float* hrow = &hprev[(size_t)mrow * HIDDIM + k + 2 * kh];
                const v2f a0 = *reinterpret_cast<const v2f*>(hrow + 0);
                const v2f a1 = *reinterpret_cast<const v2f*>(hrow + 4);
                const v2f a2 = *reinterpret_cast<const v2f*>(hrow + 8);
                const v2f a3 = *reinterpret_cast<const v2f*>(hrow + 12);
                const int p = (k >> 1) + kh;
                const v2f b0 = ldsW[(p + 0) * 16 + laneLo];
                const v2f b1 = ldsW[(p + 2) * 16 + laneLo];
                const v2f b2 = ldsW[(p + 4) * 16 + laneLo];
                const v2f b3 = ldsW[(p + 6) * 16 + laneLo];
                acc0 = wmma_f32_16x16x4(a0, b0, acc0);
                acc1 = wmma_f32_16x16x4(a1, b1, acc1);
                acc2 = wmma_f32_16x16x4(a2, b2, acc2);
                acc3 = wmma_f32_16x16x4(a3, b3, acc3);
            }
            const v8f t = (acc0 + acc1) + (acc2 + acc3);
#pragma unroll
            for (int j = 0; j < 8; ++j) {
                const size_t idx = (size_t)(rowBase + j) * HIDDIM + n0 + laneLo;
                xp[idx] = tanhf(xp[idx] + t[j]);
            }
        }

        // Device-scope step barrier (64 blocks are co-resident by construction).
        __threadfence();
        __syncthreads();
        if (tid == 0) {
            __hip_atomic_fetch_add(barrier, 1u, __ATOMIC_RELEASE,
                                   __HIP_MEMORY_SCOPE_AGENT);
            const unsigned target = (unsigned)(s + 1) * nblocks;
            while (__hip_atomic_load(barrier, __ATOMIC_ACQUIRE,
                                     __HIP_MEMORY_SCOPE_AGENT) < target) {
                __builtin_amdgcn_s_sleep(2);
            }
        }
        __syncthreads();
        __threadfence();
    }
}

// ---------------------------------------------------------------------------
extern "C" void kernel_launch(void* const* d_in, const int* in_sizes, int n_in,
                              void* d_out, int out_size, void* d_ws, size_t ws_size,
                              hipStream_t stream) {
    const float* input = (const float*)d_in[0];   // [512,64,1024]
    const float* W_ih  = (const float*)d_in[1];   // [1024,1024]
    const float* W_hh  = (const float*)d_in[2];   // [1024,1024]
    const float* b_ih  = (const float*)d_in[3];   // [1024]
    const float* b_hh  = (const float*)d_in[4];   // [1024]
    float* out = (float*)d_out;                   // [512*64,1024]
    unsigned int* barrier = (unsigned int*)d_ws;

    dim3 g1(HIDDIM / 64, (SEQ * BATCH) / 128);    // 16 x 256 blocks
    xproj_kernel<<<g1, 256, 0, stream>>>(input, W_ih, b_ih, b_hh, out, barrier);

    rnn_scan_kernel<<<dim3(HIDDIM / 16), 128, 0, stream>>>(W_hh, out, barrier);
}